// RandomlyLocalizedConformalRiskControl_4294967296360
// MI455X (gfx1250) — compile-verified
//
#include <hip/hip_runtime.h>
#include <math.h>

// ---------------------------------------------------------------------------
// RandomlyLocalizedConformalRiskControl for MI455X (gfx1250, wave32)
//
// Pipeline:
//  1) feat_kernel      : bilinear(64->256, half-pixel, clamp) + sigmoid   (288 samples)
//  2) sumsq_kernel     : per-sample ||f||^2 (deterministic LDS tree)
//  3) gram_wmma_kernel : G[32,256] = tf @ cf^T via V_WMMA_F32_16X16X4_F32 (f32-exact)
//  4) weight_kernel    : w = softmax-like RBF row normalization
//  5) cdf_scatter      : per cal sample: 8192-bin histogram over float-bit
//                        prefixes (bits>>17), exclusive scan in LDS, bucket
//                        scatter of positive-pixel score bits (ds_add_rtn)
//  6) bisect_kernel    : 50-iter bisection; exact count(score < t) =
//                        CDF[bin] + scan of one bucket (bit-compare)
//  7) final_kernel     : pred set size / test FNR
// ---------------------------------------------------------------------------

typedef __attribute__((ext_vector_type(2))) float v2f;
typedef __attribute__((ext_vector_type(8))) float v8f;

#define PIX        65536        // 256*256
#define NCAL       256
#define NTEST      32
#define NBINS      8192
#define BIN_SHIFT  17
#define CDF_STRIDE (NBINS + 1)  // 8193 entries per sample, last = pos count
#define ALPHA_F    0.1f
#define INV_2SIG2  (1.0f / 8192.0f)   // 1/(2*64*64)

// ---- workspace byte offsets (total ~144 MB) ----
#define OFF_CALRES  0ull
#define OFF_TESTRES (OFF_CALRES  + (size_t)NCAL  * PIX * 4)          // 64 MB
#define OFF_SQ      (OFF_TESTRES + (size_t)NTEST * PIX * 4)          // +8 MB
#define OFF_G       (OFF_SQ      + 2048)
#define OFF_W       (OFF_G       + (size_t)NTEST * NCAL * 4)
#define OFF_WLAST   (OFF_W       + (size_t)NTEST * NCAL * 4)
#define OFF_LAM     (OFF_WLAST   + 128)
#define OFF_POS     (OFF_LAM     + 128)
#define OFF_CDF     (OFF_POS     + 1024)
#define OFF_BKT     (OFF_CDF     + (size_t)NCAL * CDF_STRIDE * 4)    // +8.4 MB
// OFF_BKT size = NCAL*PIX*4 = 64 MB

// ---------------------------------------------------------------------------
// 1) features: half-pixel bilinear x4 upsample (edge clamp == jax resize) + sigmoid
//    grid.x = 288*256 (sample, out-row), block = 256 (out-col)
// ---------------------------------------------------------------------------
__global__ void feat_kernel(const float* __restrict__ cal,
                            const float* __restrict__ test,
                            float* __restrict__ calRes,
                            float* __restrict__ testRes) {
  int bid = blockIdx.x;
  int s  = bid >> 8;          // 0..287
  int oy = bid & 255;
  int ox = threadIdx.x;

  float fy = oy * 0.25f - 0.375f;
  float fx = ox * 0.25f - 0.375f;
  int y0 = (int)floorf(fy); float wy = fy - (float)y0;
  int x0 = (int)floorf(fx); float wx = fx - (float)x0;
  int ya = min(max(y0, 0), 63),     yb = min(max(y0 + 1, 0), 63);
  int xa = min(max(x0, 0), 63),     xb = min(max(x0 + 1, 0), 63);

  const float* src = (s < NCAL) ? (cal + (size_t)s * 4096)
                                : (test + (size_t)(s - NCAL) * 4096);
  float v00 = src[ya * 64 + xa], v01 = src[ya * 64 + xb];
  float v10 = src[yb * 64 + xa], v11 = src[yb * 64 + xb];
  float top = v00 + wx * (v01 - v00);
  float bot = v10 + wx * (v11 - v10);
  float v   = top + wy * (bot - top);
  float sig = 1.0f / (1.0f + expf(-v));   // T = 1.0

  if (s < NCAL) calRes[(size_t)s * PIX + oy * 256 + ox] = sig;
  else          testRes[(size_t)(s - NCAL) * PIX + oy * 256 + ox] = sig;
}

// ---------------------------------------------------------------------------
// 2) per-sample sum of squares. grid = 288 blocks x 256 threads.
// ---------------------------------------------------------------------------
__global__ void sumsq_kernel(const float* __restrict__ calRes,
                             const float* __restrict__ testRes,
                             float* __restrict__ sq) {
  int s = blockIdx.x, tid = threadIdx.x;
  const float* src = (s < NCAL) ? (calRes + (size_t)s * PIX)
                                : (testRes + (size_t)(s - NCAL) * PIX);
  float acc = 0.f;
  for (int p = tid; p < PIX; p += 256) { float v = src[p]; acc += v * v; }
  __shared__ float red[256];
  red[tid] = acc; __syncthreads();
  for (int st = 128; st > 0; st >>= 1) {
    if (tid < st) red[tid] += red[tid + st];
    __syncthreads();
  }
  if (tid == 0) sq[s] = red[0];
}

// ---------------------------------------------------------------------------
// 3) Gram matrix G[b,n] = sum_k tf[b,k]*cf[n,k] using f32 WMMA 16x16x4.
//    grid = 32 blocks (mTile in {0,1} x nTile in {0..15}); 8 waves split K;
//    partial C tiles combined via a fixed-order LDS sum (deterministic).
//    A fragment (16x4): lanes 0-15 -> M=lane, K={k,k+1}; lanes 16-31 -> M=lane-16, K={k+2,k+3}.
//    B fragment (4x16): mirror with N in place of M.
// ---------------------------------------------------------------------------
__global__ void gram_wmma_kernel(const float* __restrict__ testRes,
                                 const float* __restrict__ calRes,
                                 float* __restrict__ G) {
  int mTile = blockIdx.x & 1;
  int nTile = blockIdx.x >> 1;
  int tid  = threadIdx.x;
  int wave = tid >> 5, lane = tid & 31;

  int row  = (lane < 16) ? lane : (lane - 16);
  int kOff = (lane < 16) ? 0 : 2;
  const float* aBase = testRes + (size_t)(mTile * 16 + row) * PIX + kOff;
  const float* bBase = calRes  + (size_t)(nTile * 16 + row) * PIX + kOff;

  v8f c = {0.f, 0.f, 0.f, 0.f, 0.f, 0.f, 0.f, 0.f};
  int k0 = wave * (PIX / 8);
  int k1 = k0 + (PIX / 8);
#pragma unroll 8
  for (int k = k0; k < k1; k += 4) {
    v2f a, b;
    a.x = aBase[k];  a.y = aBase[k + 1];
    b.x = bBase[k];  b.y = bBase[k + 1];
    // 8 args: (neg_a, A, neg_b, B, c_mod, C, reuse_a, reuse_b)
    c = __builtin_amdgcn_wmma_f32_16x16x4_f32(false, a, false, b,
                                              (short)0, c, false, false);
  }

  __shared__ float lds[8 * 256];
#pragma unroll
  for (int r = 0; r < 8; ++r) lds[wave * 256 + r * 32 + lane] = c[r];
  __syncthreads();

  // thread i sums the 8 wave-partials in fixed order, then maps (r,lane)->(M,N)
  float s = 0.f;
#pragma unroll
  for (int w = 0; w < 8; ++w) s += lds[w * 256 + tid];
  int r = tid >> 5, l = tid & 31;
  int M = (l < 16) ? r : (r + 8);
  int N = (l < 16) ? l : (l - 16);
  G[(size_t)(mTile * 16 + M) * NCAL + nTile * 16 + N] = s;
}

// ---------------------------------------------------------------------------
// 4) Gaussian RBF weights, row-normalized with self-weight 1. 32 blocks x 256.
// ---------------------------------------------------------------------------
__global__ void weight_kernel(const float* __restrict__ G,
                              const float* __restrict__ sq,
                              float* __restrict__ w,
                              float* __restrict__ wlast) {
  int b = blockIdx.x, n = threadIdx.x;
  float d2 = sq[NCAL + b] + sq[n] - 2.0f * G[(size_t)b * NCAL + n];
  float e  = expf(-d2 * INV_2SIG2);
  __shared__ float red[256];
  red[n] = e; __syncthreads();
  for (int st = 128; st > 0; st >>= 1) {
    if (n < st) red[n] += red[n + st];
    __syncthreads();
  }
  float denom = red[0] + 1.0f;       // + self weight
  w[(size_t)b * NCAL + n] = e / denom;
  if (n == 0) wlast[b] = 1.0f / denom;
}

// ---------------------------------------------------------------------------
// 5) per cal sample: histogram of positive-pixel score bit-prefixes,
//    exclusive scan -> CDF, then bucket scatter of the raw score bits.
//    Positive f32 ordering == u32 ordering of the bit patterns.
//    grid = 256 blocks x 256 threads; LDS = 32KB bins + 1KB aux.
// ---------------------------------------------------------------------------
__global__ void cdf_scatter_kernel(const float* __restrict__ calRes,
                                   const int* __restrict__ calGt,
                                   unsigned* __restrict__ cdf,
                                   unsigned* __restrict__ pos,
                                   unsigned* __restrict__ bkt) {
  int n = blockIdx.x, tid = threadIdx.x;
  __shared__ unsigned bins[NBINS];
  __shared__ unsigned chunk[256];
  __shared__ unsigned sTotal;

  for (int i = tid; i < NBINS; i += 256) bins[i] = 0u;
  __syncthreads();

  const float* sr = calRes + (size_t)n * PIX;
  const int*   gr = calGt  + (size_t)n * PIX;

  for (int p = tid; p < PIX; p += 256) {
    if (gr[p] > 0) {
      unsigned bits = __float_as_uint(sr[p]);
      atomicAdd(&bins[bits >> BIN_SHIFT], 1u);     // ds_add_u32
    }
  }
  __syncthreads();

  // exclusive scan of bins: per-thread chunk of 32, serial top scan (cheap)
  const int CHUNK = NBINS / 256;                   // 32
  int base = tid * CHUNK;
  unsigned local = 0;
  for (int j = 0; j < CHUNK; ++j) local += bins[base + j];
  chunk[tid] = local;
  __syncthreads();
  if (tid == 0) {
    unsigned run = 0;
    for (int i = 0; i < 256; ++i) { unsigned t = chunk[i]; chunk[i] = run; run += t; }
    sTotal = run;
  }
  __syncthreads();
  {
    unsigned run = chunk[tid];
    for (int j = 0; j < CHUNK; ++j) {
      unsigned t = bins[base + j];
      bins[base + j] = run;                        // exclusive CDF
      run += t;
    }
  }
  __syncthreads();                                 // bins fully exclusive

  unsigned* cd = cdf + (size_t)n * CDF_STRIDE;
  for (int i = tid; i < NBINS; i += 256) cd[i] = bins[i];
  if (tid == 0) { cd[NBINS] = sTotal; pos[n] = sTotal; }
  __syncthreads();                                 // CDF snapshot done before scatter mutates bins

  // bucket scatter: within-bucket order is irrelevant (queries only count),
  // so ds_add_rtn-based slot assignment keeps results deterministic.
  unsigned* bp = bkt + (size_t)n * PIX;
  for (int p = tid; p < PIX; p += 256) {
    if (gr[p] > 0) {
      unsigned bits = __float_as_uint(sr[p]);
      unsigned idx = atomicAdd(&bins[bits >> BIN_SHIFT], 1u);  // ds_add_rtn_u32
      bp[idx] = bits;
    }
  }
}

// ---------------------------------------------------------------------------
// 6) 50-iteration bisection per test sample. Block b (32 blocks), thread n.
//    Exact count(score_n < t) = CDF[n][bin(t)] + bucket scan with u32 compare.
// ---------------------------------------------------------------------------
__global__ void bisect_kernel(const float* __restrict__ w,
                              const float* __restrict__ wlast,
                              const unsigned* __restrict__ pos,
                              const unsigned* __restrict__ cdf,
                              const unsigned* __restrict__ bkt,
                              float* __restrict__ lamWs,
                              float* __restrict__ out) {
  int b = blockIdx.x, n = threadIdx.x;
  float wn   = w[(size_t)b * NCAL + n];
  float posf = (float)max(pos[n], 1u);
  const unsigned* cd = cdf + (size_t)n * CDF_STRIDE;
  const unsigned* bp = bkt + (size_t)n * PIX;
  float wl = wlast[b];

  __shared__ float red[256];
  float low = 0.f, high = 1.f;

  for (int it = 0; it < 50; ++it) {
    float lam = 0.5f * (low + high);
    float t   = 1.0f - lam;                  // in (0,1) => positive float
    unsigned tb  = __float_as_uint(t);
    unsigned bin = tb >> BIN_SHIFT;          // <= 8127 < NBINS
    unsigned c0 = cd[bin], c1 = cd[bin + 1];
    unsigned cnt = c0;
    for (unsigned j = c0; j < c1; ++j)
      cnt += (bp[j] < tb) ? 1u : 0u;         // strict '<' == searchsorted left

    red[n] = wn * ((float)cnt / posf);
    __syncthreads();
    for (int st = 128; st > 0; st >>= 1) {
      if (n < st) red[n] += red[n + st];
      __syncthreads();
    }
    float risk = red[0] + wl;                // + w[:, -1] * B_CONST(=1)
    __syncthreads();
    if (risk < ALPHA_F) high = lam; else low = lam;
  }

  if (n == 0) { lamWs[b] = low; out[32 + b] = low; }
}

// ---------------------------------------------------------------------------
// 7) prediction-set size + test FNR. 32 blocks x 256 threads.
// ---------------------------------------------------------------------------
__global__ void final_kernel(const float* __restrict__ testRes,
                             const int* __restrict__ testGt,
                             const float* __restrict__ lamWs,
                             float* __restrict__ out) {
  int b = blockIdx.x, tid = threadIdx.x;
  float t = 1.0f - lamWs[b];
  const float* sr = testRes + (size_t)b * PIX;
  const int*   gr = testGt  + (size_t)b * PIX;
  unsigned sz = 0, tp = 0, tpos = 0;
  for (int p = tid; p < PIX; p += 256) {
    bool pr = sr[p] >= t;
    bool g  = gr[p] > 0;
    sz   += pr ? 1u : 0u;
    tp   += (pr && g) ? 1u : 0u;
    tpos += g ? 1u : 0u;
  }
  __shared__ unsigned r0[256], r1[256], r2[256];
  r0[tid] = sz; r1[tid] = tp; r2[tid] = tpos; __syncthreads();
  for (int st = 128; st > 0; st >>= 1) {
    if (tid < st) { r0[tid] += r0[tid + st]; r1[tid] += r1[tid + st]; r2[tid] += r2[tid + st]; }
    __syncthreads();
  }
  if (tid == 0) {
    float tposf = (float)max(r2[0], 1u);
    out[b]      = 1.0f - (float)r1[0] / tposf;   // fnr_test
    out[64 + b] = (float)r0[0];                  // size
  }
}

// ---------------------------------------------------------------------------
extern "C" void kernel_launch(void* const* d_in, const int* in_sizes, int n_in,
                              void* d_out, int out_size, void* d_ws, size_t ws_size,
                              hipStream_t stream) {
  (void)in_sizes; (void)n_in; (void)out_size; (void)ws_size;
  const float* cal_feat  = (const float*)d_in[0];  // [256,1,64,64]
  const float* test_feat = (const float*)d_in[1];  // [32,1,64,64]
  const int*   cal_gt    = (const int*)d_in[2];    // [256,1,256,256]
  const int*   test_gt   = (const int*)d_in[3];    // [32,1,256,256]

  char* ws = (char*)d_ws;
  float*    calRes  = (float*)(ws + OFF_CALRES);
  float*    testRes = (float*)(ws + OFF_TESTRES);
  float*    sq      = (float*)(ws + OFF_SQ);
  float*    G       = (float*)(ws + OFF_G);
  float*    W       = (float*)(ws + OFF_W);
  float*    WLAST   = (float*)(ws + OFF_WLAST);
  float*    LAM     = (float*)(ws + OFF_LAM);
  unsigned* POS     = (unsigned*)(ws + OFF_POS);
  unsigned* CDF     = (unsigned*)(ws + OFF_CDF);
  unsigned* BKT     = (unsigned*)(ws + OFF_BKT);
  float*    out     = (float*)d_out;               // [fnr(32) | lam(32) | size(32)]

  feat_kernel<<<288 * 256, 256, 0, stream>>>(cal_feat, test_feat, calRes, testRes);
  sumsq_kernel<<<288, 256, 0, stream>>>(calRes, testRes, sq);
  gram_wmma_kernel<<<32, 256, 0, stream>>>(testRes, calRes, G);
  weight_kernel<<<32, 256, 0, stream>>>(G, sq, W, WLAST);
  cdf_scatter_kernel<<<256, 256, 0, stream>>>(calRes, cal_gt, CDF, POS, BKT);
  bisect_kernel<<<32, 256, 0, stream>>>(W, WLAST, POS, CDF, BKT, LAM, out);
  final_kernel<<<32, 256, 0, stream>>>(testRes, test_gt, LAM, out);
}